// GT2Center_5214090297417
// MI455X (gfx1250) — compile-verified
//
#include <hip/hip_runtime.h>
#include <hip/hip_bf16.h>

typedef float v2f __attribute__((ext_vector_type(2)));
typedef float v8f __attribute__((ext_vector_type(8)));

#define BATCH      16
#define CHANS      256
#define NPIX       16384        // 128*128
#define KCLS       6
#define KPAD       8
#define IGNORE_IDX 255

#define NCHUNK     256          // pixels staged per LDS tile
#define ROWSTRIDE  260          // 260 % 64 == 4 -> conflict-free b64 A-fragment reads
#define WAVES      4
#define THREADS    (WAVES * 32)
#define NSPLIT     8            // N-dimension splits per (b, cTile)
#define STEPS      (NCHUNK / (WAVES * 4))   // 16 WMMA steps per wave per tile

// ---------------------------------------------------------------------------
// Zero the accumulator region of the workspace (graph-replay safe re-init).
// ---------------------------------------------------------------------------
__global__ void zero_acc_kernel(float* __restrict__ acc, int n) {
    int i = blockIdx.x * blockDim.x + threadIdx.x;
    if (i < n) acc[i] = 0.0f;
}

// ---------------------------------------------------------------------------
// Per-batch class histogram (counts), written as float into ws.
// ---------------------------------------------------------------------------
__global__ void count_kernel(const int* __restrict__ gt, float* __restrict__ counts) {
    __shared__ int lc[KPAD];
    int b = blockIdx.x;
    if (threadIdx.x < KPAD) lc[threadIdx.x] = 0;
    __syncthreads();
    const int* gb = gt + (size_t)b * NPIX;
    for (int i = threadIdx.x; i < NPIX; i += blockDim.x) {
        int l = gb[i];
        if (l >= 0 && l < KCLS) atomicAdd(&lc[l], 1);   // IGNORE (255) excluded
    }
    __syncthreads();
    if (threadIdx.x < KPAD) counts[b * KPAD + threadIdx.x] = (float)lc[threadIdx.x];
}

// ---------------------------------------------------------------------------
// Main pass: D[c,k] += x[c,n] * onehot(label[n], k) via v_wmma_f32_16x16x4_f32.
// Block = (b, cTile, split). 4 waves share one LDS x-tile; each wave owns two
// ping-pong 16x16 accumulators and handles every 4th K-step of the tile.
// ---------------------------------------------------------------------------
__global__ void __launch_bounds__(THREADS)
center_accum_kernel(const float* __restrict__ x, const int* __restrict__ gt,
                    float* __restrict__ acc /* [B][CHANS][KPAD] */) {
    __shared__ float tile[16 * ROWSTRIDE];
    __shared__ int   labs[NCHUNK];

    int wg    = blockIdx.x;
    int split = wg % NSPLIT;
    int ct    = (wg / NSPLIT) % (CHANS / 16);
    int b     = wg / (NSPLIT * (CHANS / 16));

    const int nPerSplit = NPIX / NSPLIT;          // 2048
    int nBeg = split * nPerSplit;
    int nEnd = nBeg + nPerSplit;

    const float* xb = x  + ((size_t)b * CHANS + (size_t)ct * 16) * NPIX;
    const int*   gb = gt + (size_t)b * NPIX;

    int tid  = threadIdx.x;
    int lane = tid & 31;
    int wave = tid >> 5;
    int m    = lane & 15;      // A-row (channel) / B-col (class) / D-col (class)
    int hi   = lane >> 4;      // lane half selects K-subgroup and D row-half

    v8f d0 = {};               // D[c = j + 8*hi][k = m], even steps
    v8f d1 = {};               // odd steps (breaks WMMA->WMMA RAW chain)

    for (int n0 = nBeg; n0 < nEnd; n0 += NCHUNK) {
        // Stage x tile: 16 channels x NCHUNK pixels, 16B-vectorized, coalesced.
        {
            const float4* src = (const float4*)(xb + n0);   // row 0 base
            for (int i = tid; i < 16 * (NCHUNK / 4); i += THREADS) {
                int r  = i / (NCHUNK / 4);
                int p4 = i % (NCHUNK / 4);
                float4 v = src[(size_t)r * (NPIX / 4) + p4];
                *(float4*)&tile[r * ROWSTRIDE + p4 * 4] = v;
            }
            if (tid < NCHUNK / 4)
                ((int4*)labs)[tid] = ((const int4*)(gb + n0))[tid];
        }
        __syncthreads();

        // Fully unrolled: wave w handles K-steps s = w*4 + it*16, it = 0..15.
#pragma unroll
        for (int it = 0; it < STEPS; ++it) {
            int s = wave * 4 + it * (WAVES * 4);
            // A fragment (16x4 f32): lane -> row m, cols s+2*hi, s+2*hi+1
            const float* ap = &tile[m * ROWSTRIDE + s + 2 * hi];
            v2f a; a.x = ap[0]; a.y = ap[1];
            // B fragment (4x16 f32 one-hot): VGPR j -> row s+j+2*hi, col m
            int l0 = labs[s + 2 * hi];
            int l1 = labs[s + 2 * hi + 1];
            v2f bm;
            bm.x = (l0 == m) ? 1.0f : 0.0f;
            bm.y = (l1 == m) ? 1.0f : 0.0f;
            if (it & 1)
                d1 = __builtin_amdgcn_wmma_f32_16x16x4_f32(
                        false, a, false, bm, (short)0, d1, false, false);
            else
                d0 = __builtin_amdgcn_wmma_f32_16x16x4_f32(
                        false, a, false, bm, (short)0, d0, false, false);
        }
        __syncthreads();
    }

    // Epilogue: lane (m<6) holds D[c=j+8*hi][k=m] in d0[j]+d1[j]; scatter-add.
    if (m < KCLS) {
        float* accb = acc + ((size_t)b * CHANS + (size_t)ct * 16) * KPAD;
#pragma unroll
        for (int j = 0; j < 8; ++j) {
            int c = j + 8 * hi;
            atomicAdd(&accb[c * KPAD + m], d0[j] + d1[j]);
        }
    }
}

// ---------------------------------------------------------------------------
// Finalize: out[b][k][c] = acc[b][c][k] / max(count[b][k], 1)
// ---------------------------------------------------------------------------
__global__ void finalize_kernel(const float* __restrict__ acc,
                                const float* __restrict__ counts,
                                float* __restrict__ out) {
    int idx = blockIdx.x * blockDim.x + threadIdx.x;
    if (idx >= BATCH * KCLS * CHANS) return;
    int c = idx % CHANS;
    int k = (idx / CHANS) % KCLS;
    int b = idx / (CHANS * KCLS);
    float cnt = fmaxf(counts[b * KPAD + k], 1.0f);
    out[idx] = acc[((size_t)b * CHANS + c) * KPAD + k] / cnt;
}

extern "C" void kernel_launch(void* const* d_in, const int* in_sizes, int n_in,
                              void* d_out, int out_size, void* d_ws, size_t ws_size,
                              hipStream_t stream) {
    const float* x  = (const float*)d_in[0];
    const int*   gt = (const int*)d_in[1];
    float*       out = (float*)d_out;

    // Workspace layout: [acc: B*CHANS*KPAD floats][counts: B*KPAD floats]
    float* acc    = (float*)d_ws;
    float* counts = acc + (size_t)BATCH * CHANS * KPAD;

    const int accN = BATCH * CHANS * KPAD;                    // 32768
    zero_acc_kernel<<<(accN + 255) / 256, 256, 0, stream>>>(acc, accN);

    count_kernel<<<BATCH, 256, 0, stream>>>(gt, counts);

    const int nBlocks = BATCH * (CHANS / 16) * NSPLIT;        // 2048
    center_accum_kernel<<<nBlocks, THREADS, 0, stream>>>(x, gt, acc);

    const int outN = BATCH * KCLS * CHANS;                    // 24576
    finalize_kernel<<<(outN + 255) / 256, 256, 0, stream>>>(acc, counts, out);
}